// GraphLayerLinearNRI_51101520888316
// MI455X (gfx1250) — compile-verified
//
#include <hip/hip_runtime.h>

// Problem constants (from reference): N=16, K=128, CIN=64, C=64, ET=2
#define GN 16
#define GK 128
#define GC 64
#define ROWS (GN * GK)          // 2048 flattened node rows

typedef float v2f __attribute__((ext_vector_type(2)));
typedef float v8f __attribute__((ext_vector_type(8)));

// ---------------------------------------------------------------------------
// f32 WMMA 16x16x4 wrapper (CDNA5): D = A(16x4) * B(4x16) + C(16x16)
// ---------------------------------------------------------------------------
__device__ __forceinline__ v8f wmma_f32(v2f a, v2f b, v8f c) {
  return __builtin_amdgcn_wmma_f32_16x16x4_f32(false, a, false, b, (short)0, c,
                                               false, false);
}

// A fragment: 16x4 f32 tile at (rows m=0..15, cols 4*kc..4*kc+3), row-major ld.
// ISA layout: lanes 0-15 hold K=0,1 (v0,v1); lanes 16-31 hold K=2,3.
__device__ __forceinline__ v2f load_afrag(const float* A, int ld, int kc,
                                          int lane) {
  const int m = lane & 15;
  const int koff = (lane >> 4) * 2;
  const float* p = A + m * ld + kc * 4 + koff;
  v2f r;
  r.x = p[0];
  r.y = p[1];
  return r;
}

// B fragment from pre-packed weights: one coalesced 8-byte load per fragment.
// Pack order: Wp[((kc*4 + nt)*32 + lane)*2 + {0,1}]
__device__ __forceinline__ v2f load_bfrag_packed(const float* Wp, int kc,
                                                 int nt, int lane) {
  return *(const v2f*)(Wp + (size_t)(((kc * 4 + nt) << 5) + lane) * 2);
}

// Preload the 32 A fragments of a 32x64 slab (2 M-tiles x 16 k-chunks).
__device__ __forceinline__ void load_afrags_32(const float* A, int lda,
                                               int lane, v2f af[16][2]) {
#pragma unroll
  for (int kc = 0; kc < 16; ++kc) {
    af[kc][0] = load_afrag(A, lda, kc, lane);
    af[kc][1] = load_afrag(A + 16 * lda, lda, kc, lane);
  }
}

// GEMM with A resident in registers; B double-buffered one k-step ahead so
// the wmma stream never drains loadcnt to zero.
__device__ __forceinline__ void gemm_regA(const v2f af[16][2], const float* Wp,
                                          int lane, v8f acc[2][4]) {
  v2f bb[2][4];
#pragma unroll
  for (int nt = 0; nt < 4; ++nt) bb[0][nt] = load_bfrag_packed(Wp, 0, nt, lane);
#pragma unroll
  for (int kc = 0; kc < 16; ++kc) {
    const int cur = kc & 1, nxt = cur ^ 1;
    if (kc < 15) {
#pragma unroll
      for (int nt = 0; nt < 4; ++nt)
        bb[nxt][nt] = load_bfrag_packed(Wp, kc + 1, nt, lane);
    }
#pragma unroll
    for (int nt = 0; nt < 4; ++nt) {
      acc[0][nt] = wmma_f32(af[kc][0], bb[cur][nt], acc[0][nt]);
      acc[1][nt] = wmma_f32(af[kc][1], bb[cur][nt], acc[1][nt]);
    }
  }
}

// GEMM reading A from memory per k-chunk, B pipelined as above.
__device__ __forceinline__ void gemm_memA(const float* A, int lda,
                                          const float* Wp, int lane,
                                          v8f acc[2][4]) {
  v2f bb[2][4];
#pragma unroll
  for (int nt = 0; nt < 4; ++nt) bb[0][nt] = load_bfrag_packed(Wp, 0, nt, lane);
#pragma unroll
  for (int kc = 0; kc < 16; ++kc) {
    const int cur = kc & 1, nxt = cur ^ 1;
    v2f a0 = load_afrag(A, lda, kc, lane);
    v2f a1 = load_afrag(A + 16 * lda, lda, kc, lane);
    if (kc < 15) {
#pragma unroll
      for (int nt = 0; nt < 4; ++nt)
        bb[nxt][nt] = load_bfrag_packed(Wp, kc + 1, nt, lane);
    }
#pragma unroll
    for (int nt = 0; nt < 4; ++nt) {
      acc[0][nt] = wmma_f32(a0, bb[cur][nt], acc[0][nt]);
      acc[1][nt] = wmma_f32(a1, bb[cur][nt], acc[1][nt]);
    }
  }
}

// Store one 16x16 D tile (row-major dest, leading dim ld, column offset).
// C/D layout: VGPR g -> M = g (lanes 0-15) or g+8 (lanes 16-31); N = lane&15.
__device__ __forceinline__ void store_tile(float* D, int ld, int coloff,
                                           int lane, v8f d) {
  const int n = (lane & 15) + coloff;
  const int mbase = (lane >> 4) * 8;
#pragma unroll
  for (int g = 0; g < 8; ++g) D[(mbase + g) * ld + n] = d[g];
}

__device__ __forceinline__ void zero_acc(v8f acc[2][4]) {
#pragma unroll
  for (int mt = 0; mt < 2; ++mt)
#pragma unroll
    for (int nt = 0; nt < 4; ++nt) {
      v8f z = {0.f, 0.f, 0.f, 0.f, 0.f, 0.f, 0.f, 0.f};
      acc[mt][nt] = z;
    }
}

__device__ __forceinline__ float fast_tanh(float x) {
#if __has_builtin(__builtin_amdgcn_tanhf)
  return __builtin_amdgcn_tanhf(x);  // v_tanh_f32 (gfx1250 TRANS op)
#else
  float e = __expf(-2.0f * __builtin_fabsf(x));
  float r = (1.0f - e) / (1.0f + e);
  return __builtin_copysignf(r, x);
#endif
}

// ---------------------------------------------------------------------------
// Kernel 0: pack the nine 64x64 weight matrices into B-fragment order.
// Matrices: 0 W_lin[1:], 1 W_u1, 2 W_u2, 3..6 W_b1[e]{top,bot}, 7..8 W_b2[e]
// ---------------------------------------------------------------------------
__global__ __launch_bounds__(256) void k0_pack(
    const float* __restrict__ W_lin, const float* __restrict__ W_u1,
    const float* __restrict__ W_u2, const float* __restrict__ W_b1,
    const float* __restrict__ W_b2, float* __restrict__ Wpack) {
  const int b = blockIdx.x;
  const float* src = W_u1;
  if (b == 0) src = W_lin + 64;
  else if (b == 1) src = W_u1;
  else if (b == 2) src = W_u2;
  else if (b == 3) src = W_b1;                  // e0 rows 0..63
  else if (b == 4) src = W_b1 + 64 * 64;        // e0 rows 64..127
  else if (b == 5) src = W_b1 + 128 * 64;       // e1 rows 0..63
  else if (b == 6) src = W_b1 + 192 * 64;       // e1 rows 64..127
  else if (b == 7) src = W_b2;                  // e0
  else src = W_b2 + 64 * 64;                    // e1
  float* dst = Wpack + (size_t)b * 4096;
  for (int idx = threadIdx.x; idx < 4096; idx += 256) {
    const int p = idx >> 1, el = idx & 1;
    const int lane = p & 31;
    const int ntkc = p >> 5;
    const int nt = ntkc & 3;
    const int kc = ntkc >> 2;
    const int koff = (lane >> 4) * 2;
    const int n = (lane & 15) + nt * 16;
    const int k = kc * 4 + koff + el;
    dst[idx] = src[k * 64 + n];
  }
}

// ---------------------------------------------------------------------------
// Kernel 1: per 32-row slab (one wave):
//   h     = x @ W_lin[1:] + (b_lin + t*W_lin[0])        (kept in LDS)
//   unary = tanh(h@W_u1 + b_u1) @ W_u2 + b_u2           -> global
//   A[e]  = h @ W_b1[e][0:64]  + b_b1[e]                -> global
//   B[e]  = h @ W_b1[e][64:128]                         -> global
// The h fragments are loaded into registers ONCE and reused by 5 GEMMs.
// grid: 8 blocks x 256 threads
// ---------------------------------------------------------------------------
__global__ __launch_bounds__(256) void k1_precompute(
    const float* __restrict__ t, const float* __restrict__ x,
    const float* __restrict__ W_lin, const float* __restrict__ b_lin,
    const float* __restrict__ b_u1, const float* __restrict__ b_u2,
    const float* __restrict__ b_b1, const float* __restrict__ Wpack,
    float* __restrict__ unary, float* __restrict__ Abuf,
    float* __restrict__ Bbuf) {
  __shared__ float lds_h[8][32 * 64];
  __shared__ float lds_t[8][32 * 64];
  const int tid = threadIdx.x;
  const int wave = tid >> 5;
  const int lane = tid & 31;
  const int row0 = blockIdx.x * 256 + wave * 32;
  const float tval = t[0];
  const int col = lane & 15;

  v8f acc[2][4];

  // ---- h = x @ W_lin[1:] + (b_lin + t*W_lin[0]) ----
  zero_acc(acc);
  gemm_memA(x + (size_t)row0 * 64, 64, Wpack /*lin*/, lane, acc);
#pragma unroll
  for (int mt = 0; mt < 2; ++mt)
#pragma unroll
    for (int nt = 0; nt < 4; ++nt) {
      const int c = col + nt * 16;
      const float bb = b_lin[c] + tval * W_lin[c];
#pragma unroll
      for (int g = 0; g < 8; ++g) acc[mt][nt][g] += bb;
      store_tile(lds_h[wave] + mt * 16 * 64, 64, nt * 16, lane, acc[mt][nt]);
    }

  // ---- resident A fragments of h (reused by the next 5 GEMMs) ----
  v2f afrH[16][2];
  load_afrags_32(lds_h[wave], 64, lane, afrH);

  // ---- unary: hid = tanh(h@W_u1 + b_u1); unary = hid@W_u2 + b_u2 ----
  zero_acc(acc);
  gemm_regA(afrH, Wpack + 4096 /*u1*/, lane, acc);
#pragma unroll
  for (int mt = 0; mt < 2; ++mt)
#pragma unroll
    for (int nt = 0; nt < 4; ++nt) {
      const float bb = b_u1[col + nt * 16];
#pragma unroll
      for (int g = 0; g < 8; ++g) acc[mt][nt][g] = fast_tanh(acc[mt][nt][g] + bb);
      store_tile(lds_t[wave] + mt * 16 * 64, 64, nt * 16, lane, acc[mt][nt]);
    }
  zero_acc(acc);
  gemm_memA(lds_t[wave], 64, Wpack + 2 * 4096 /*u2*/, lane, acc);
#pragma unroll
  for (int mt = 0; mt < 2; ++mt)
#pragma unroll
    for (int nt = 0; nt < 4; ++nt) {
      const float bb = b_u2[col + nt * 16];
#pragma unroll
      for (int g = 0; g < 8; ++g) acc[mt][nt][g] += bb;
      store_tile(unary + (size_t)(row0 + mt * 16) * 64, 64, nt * 16, lane,
                 acc[mt][nt]);
    }

  // ---- A[e] (b_b1 folded in), B[e] ----
#pragma unroll
  for (int e = 0; e < 2; ++e) {
    zero_acc(acc);
    gemm_regA(afrH, Wpack + (size_t)(3 + 2 * e) * 4096, lane, acc);
#pragma unroll
    for (int mt = 0; mt < 2; ++mt)
#pragma unroll
      for (int nt = 0; nt < 4; ++nt) {
        const float bb = b_b1[e * 64 + col + nt * 16];
#pragma unroll
        for (int g = 0; g < 8; ++g) acc[mt][nt][g] += bb;
        store_tile(Abuf + ((size_t)e * ROWS + row0 + mt * 16) * 64, 64,
                   nt * 16, lane, acc[mt][nt]);
      }

    zero_acc(acc);
    gemm_regA(afrH, Wpack + (size_t)(4 + 2 * e) * 4096, lane, acc);
#pragma unroll
    for (int mt = 0; mt < 2; ++mt)
#pragma unroll
      for (int nt = 0; nt < 4; ++nt)
        store_tile(Bbuf + ((size_t)e * ROWS + row0 + mt * 16) * 64, 64,
                   nt * 16, lane, acc[mt][nt]);
  }
}

// ---------------------------------------------------------------------------
// Kernel 2: S[e,n,j,c] = sum_i E[n,i,j,e] * tanh(A[e,n,j,c] + B[e,n,i,c])
//           Esum[e,n,j] = sum_i E[n,i,j,e]
// grid: (16 j-groups, 16 n) = 256 blocks x 256 threads.
// Thread owns (j, 2 channels, both edge types). LDS: B 64KB + E 8KB.
// ---------------------------------------------------------------------------
__global__ __launch_bounds__(256) void k2_messages(
    const float* __restrict__ E, const float* __restrict__ Abuf,
    const float* __restrict__ Bbuf, float* __restrict__ S,
    float* __restrict__ Esum) {
  __shared__ float ldsB0[128 * 64];
  __shared__ float ldsB1[128 * 64];
  __shared__ float ldsE[128 * 16];  // [i][j'*2 + e], j' in 0..7
  const int n = blockIdx.y;
  const int jg = blockIdx.x;
  const int tid = threadIdx.x;

  // ---- cooperative staging, float4-wide ----
  {
    const float4* gB0 =
        (const float4*)(Bbuf + ((size_t)0 * ROWS + n * 128) * 64);
    const float4* gB1 =
        (const float4*)(Bbuf + ((size_t)1 * ROWS + n * 128) * 64);
    float4* sB0 = (float4*)ldsB0;
    float4* sB1 = (float4*)ldsB1;
    for (int q = tid; q < 128 * 16; q += 256) {
      sB0[q] = gB0[q];
      sB1[q] = gB1[q];
    }
    float4* sE = (float4*)ldsE;
    const size_t ebase = (size_t)n * GK * GK * 2;
    for (int q = tid; q < 128 * 4; q += 256) {
      const int i = q >> 2;
      const int je4 = (q & 3) * 4;
      sE[q] = *(const float4*)(E + ebase + (size_t)i * 256 + jg * 16 + je4);
    }
  }
  __syncthreads();

  const int jp = tid >> 5;            // 0..7
  const int j = jg * 8 + jp;          // 0..127
  const int c0 = (tid & 31) * 2;      // channel pair
  const float* Ap0 = Abuf + ((size_t)0 * ROWS + n * 128 + j) * 64 + c0;
  const float* Ap1 = Abuf + ((size_t)1 * ROWS + n * 128 + j) * 64 + c0;
  const float a00 = Ap0[0], a01 = Ap0[1];
  const float a10 = Ap1[0], a11 = Ap1[1];
  float s00 = 0.f, s01 = 0.f, s10 = 0.f, s11 = 0.f;
  float es0 = 0.f, es1 = 0.f;

#pragma unroll 4
  for (int i = 0; i < 128; ++i) {
    const float w0 = ldsE[i * 16 + jp * 2 + 0];
    const float w1 = ldsE[i * 16 + jp * 2 + 1];
    es0 += w0;
    es1 += w1;
    const float2 b0 = *(const float2*)&ldsB0[i * 64 + c0];
    const float2 b1 = *(const float2*)&ldsB1[i * 64 + c0];
    s00 += w0 * fast_tanh(a00 + b0.x);
    s01 += w0 * fast_tanh(a01 + b0.y);
    s10 += w1 * fast_tanh(a10 + b1.x);
    s11 += w1 * fast_tanh(a11 + b1.y);
  }

  *(float2*)(S + ((size_t)0 * ROWS + n * 128 + j) * 64 + c0) =
      make_float2(s00, s01);
  *(float2*)(S + ((size_t)1 * ROWS + n * 128 + j) * 64 + c0) =
      make_float2(s10, s11);
  if ((tid & 31) == 0) {
    Esum[(size_t)0 * ROWS + n * 128 + j] = es0;
    Esum[(size_t)1 * ROWS + n * 128 + j] = es1;
  }
}

// ---------------------------------------------------------------------------
// Kernel 3: out = unary + (sum_e S[e]@W_b2[e] + Esum[e]*b_b2[e]) / K
// grid: 8 blocks x 256 threads (8 waves, one 32-row slab each)
// ---------------------------------------------------------------------------
__global__ __launch_bounds__(256) void k3_output(
    const float* __restrict__ S, const float* __restrict__ Wpack,
    const float* __restrict__ b_b2, const float* __restrict__ Esum,
    const float* __restrict__ unary, float* __restrict__ out) {
  const int tid = threadIdx.x;
  const int wave = tid >> 5;
  const int lane = tid & 31;
  const int row0 = blockIdx.x * 256 + wave * 32;

  v8f acc[2][4];
  zero_acc(acc);
  gemm_memA(S + (size_t)row0 * 64, 64, Wpack + (size_t)7 * 4096, lane, acc);
  gemm_memA(S + ((size_t)ROWS + row0) * 64, 64, Wpack + (size_t)8 * 4096,
            lane, acc);

  const float inv_k = 1.0f / (float)GK;
  const int mbase = (lane >> 4) * 8;
#pragma unroll
  for (int mt = 0; mt < 2; ++mt)
#pragma unroll
    for (int nt = 0; nt < 4; ++nt) {
      const int colc = (lane & 15) + nt * 16;
#pragma unroll
      for (int g = 0; g < 8; ++g) {
        const int row = row0 + mt * 16 + mbase + g;
        float v = acc[mt][nt][g] + Esum[row] * b_b2[colc] +
                  Esum[ROWS + row] * b_b2[64 + colc];
        out[(size_t)row * 64 + colc] =
            v * inv_k + unary[(size_t)row * 64 + colc];
      }
    }
}

// ---------------------------------------------------------------------------
// Host launcher
// ---------------------------------------------------------------------------
extern "C" void kernel_launch(void* const* d_in, const int* in_sizes, int n_in,
                              void* d_out, int out_size, void* d_ws,
                              size_t ws_size, hipStream_t stream) {
  (void)in_sizes;
  (void)n_in;
  (void)out_size;
  (void)ws_size;
  const float* t = (const float*)d_in[0];
  const float* x = (const float*)d_in[1];
  const float* E = (const float*)d_in[2];
  const float* W_lin = (const float*)d_in[3];
  const float* b_lin = (const float*)d_in[4];
  const float* W_u1 = (const float*)d_in[5];
  const float* b_u1 = (const float*)d_in[6];
  const float* W_u2 = (const float*)d_in[7];
  const float* b_u2 = (const float*)d_in[8];
  const float* W_b1 = (const float*)d_in[9];
  const float* b_b1 = (const float*)d_in[10];
  const float* W_b2 = (const float*)d_in[11];
  const float* b_b2 = (const float*)d_in[12];
  float* out = (float*)d_out;

  // Workspace layout (floats): unary | A[2] | B[2] | S[2] | Esum[2] | Wpack[9]
  float* ws = (float*)d_ws;
  float* unary = ws;                      // 2048*64
  float* Abuf = unary + ROWS * 64;        // 2*2048*64
  float* Bbuf = Abuf + 2 * ROWS * 64;     // 2*2048*64
  float* Sbuf = Bbuf + 2 * ROWS * 64;     // 2*2048*64
  float* Esum = Sbuf + 2 * ROWS * 64;     // 2*2048
  float* Wpack = Esum + 2 * ROWS;         // 9*4096
  // total = 958,464 floats ~= 3.74 MB

  k0_pack<<<dim3(9), dim3(256), 0, stream>>>(W_lin, W_u1, W_u2, W_b1, W_b2,
                                             Wpack);
  k1_precompute<<<dim3(8), dim3(256), 0, stream>>>(
      t, x, W_lin, b_lin, b_u1, b_u2, b_b1, Wpack, unary, Abuf, Bbuf);
  k2_messages<<<dim3(16, 16), dim3(256), 0, stream>>>(E, Abuf, Bbuf, Sbuf,
                                                      Esum);
  k3_output<<<dim3(8), dim3(256), 0, stream>>>(Sbuf, Wpack, b_b2, Esum, unary,
                                               out);
}